// RateDistortionLoss_52106543235686
// MI455X (gfx1250) — compile-verified
//
#include <hip/hip_runtime.h>
#include <hip/hip_bf16.h>

typedef __attribute__((ext_vector_type(2))) float v2f;
typedef __attribute__((ext_vector_type(8))) float v8f;

#define WAVES_PER_BLOCK 4
#define REG_H 26   // 16 + 2*5 halo rows
#define REG_W 28   // 16 + 2*5 halo cols, padded to 28 (7 K-chunks of 4)
#define NQ 5       // x, y, x*x, y*y, x*y

// Tile bookkeeping: N=16, C=3, base 512x512; scales s=0..4, H_s=512>>s.
// Tiles per scale: 48*(32>>s)^2 -> 49152, 12288, 3072, 768, 192 ; total 65472.
#define TILES_TOTAL 65472
#define TILES_S0 49152

#define GTAB_BIAS 16   // gaussian taps live at [16..26] of the 48-slot LDS table
#define GTAB_LEN 48

// Normalized 1D Gaussian, sigma=1.5, 11 taps (separable == reference 2D outer product)
#define G0 0.00102838f
#define G1 0.00759876f
#define G2 0.03600075f
#define G3 0.10936145f
#define G4 0.21300536f
#define G5 0.26601261f

__constant__ float c_g[11] = {G0, G1, G2, G3, G4, G5, G4, G3, G2, G1, G0};

__global__ __launch_bounds__(WAVES_PER_BLOCK * 32)
void ssim_tile_kernel(const float* __restrict__ pred,
                      const float* __restrict__ tgt,
                      float* __restrict__ ssim_part,
                      float* __restrict__ mse_part) {
  __shared__ float sX[WAVES_PER_BLOCK][REG_H][REG_W];
  __shared__ float sY[WAVES_PER_BLOCK][REG_H][REG_W];
  __shared__ float sP[WAVES_PER_BLOCK][NQ][16][REG_W];
  // Pair table for band-matrix B fragments: sGP[i] = (g(i), g(i+1)),
  // g(t) = gaussian tap t-GTAB_BIAS if in [0,10], else 0. 8-byte slots ->
  // every fragment fetch is one aligned ds_load_b64, no branches, no clamps.
  __shared__ float2 sGP[GTAB_LEN];

  const int lane = threadIdx.x & 31;
  const int wv = threadIdx.x >> 5;
  const int tile = blockIdx.x * WAVES_PER_BLOCK + wv;
  const bool active = (tile < TILES_TOTAL);

  {
    const int t = threadIdx.x;
    if (t < GTAB_LEN) {
      float vx = 0.f, vy = 0.f;
      if (t >= GTAB_BIAS && t <= GTAB_BIAS + 10) vx = c_g[t - GTAB_BIAS];
      const int t2 = t + 1;
      if (t2 >= GTAB_BIAS && t2 <= GTAB_BIAS + 10) vy = c_g[t2 - GTAB_BIAS];
      sGP[t] = make_float2(vx, vy);
    }
  }

  int s = 0, local = 0;
  int ty = 0, tx = 0;

  if (active) {
    if (tile < 49152) { s = 0; local = tile; }
    else if (tile < 61440) { s = 1; local = tile - 49152; }
    else if (tile < 64512) { s = 2; local = tile - 61440; }
    else if (tile < 65280) { s = 3; local = tile - 64512; }
    else { s = 4; local = tile - 65280; }

    const int tpr = 32 >> s;        // tiles per row at this scale
    const int tpi = tpr * tpr;      // tiles per (n,c) plane
    const int img = local / tpi;    // fused n*3+c index, 0..47
    const int rem = local - img * tpi;
    ty = rem / tpr;
    tx = rem - ty * tpr;
    const int Hs = 512 >> s;
    const int bs = 1 << s;
    const float* pbase = pred + (size_t)img * (512 * 512);
    const float* tbase = tgt + (size_t)img * (512 * 512);
    const float inv_area = 1.0f / (float)(bs * bs);

    // Load zero-padded (pooled-on-the-fly) region into LDS
    for (int idx = lane; idx < REG_H * REG_W; idx += 32) {
      const int m = idx / REG_W;
      const int j = idx - m * REG_W;
      const int gr = ty * 16 + m - 5;
      const int gc = tx * 16 + j - 5;
      float xv = 0.f, yv = 0.f;
      if (gr >= 0 && gr < Hs && gc >= 0 && gc < Hs) {
        if (s == 0) {
          const size_t o = (size_t)gr * 512 + gc;
          xv = pbase[o];
          yv = tbase[o];
        } else {
          float ax = 0.f, ay = 0.f;
          const int r0 = gr * bs, c0 = gc * bs;
          for (int dy = 0; dy < bs; ++dy) {
            const size_t ro = (size_t)(r0 + dy) * 512 + c0;
            for (int dx = 0; dx < bs; ++dx) {
              ax += pbase[ro + dx];
              ay += tbase[ro + dx];
            }
          }
          xv = ax * inv_area;
          yv = ay * inv_area;
        }
      }
      sX[wv][m][j] = xv;
      sY[wv][m][j] = yv;
    }
  }
  __syncthreads();

  if (active) {
    // Vertical 11-tap pass on VALU (weights fold to immediates)
    constexpr float KG[11] = {G0, G1, G2, G3, G4, G5, G4, G3, G2, G1, G0};
    for (int idx = lane; idx < 16 * REG_W; idx += 32) {
      const int m = idx / REG_W;
      const int j = idx - m * REG_W;
      float amx = 0.f, amy = 0.f, axx = 0.f, ayy = 0.f, axy = 0.f;
#pragma unroll
      for (int k = 0; k < 11; ++k) {
        const float w = KG[k];
        const float x = sX[wv][m + k][j];
        const float y = sY[wv][m + k][j];
        amx += w * x;
        amy += w * y;
        axx += w * x * x;
        ayy += w * y * y;
        axy += w * x * y;
      }
      sP[wv][0][m][j] = amx;
      sP[wv][1][m][j] = amy;
      sP[wv][2][m][j] = axx;
      sP[wv][3][m][j] = ayy;
      sP[wv][4][m][j] = axy;
    }
  }
  __syncthreads();

  if (active) {
    // Horizontal pass via V_WMMA_F32_16X16X4_F32:
    //   O(16x16) = P(16x28) x Gh(28x16), Gh[j][n] = g[j-n] for 0<=j-n<=10
    // A layout (16x4 f32): lanes 0-15 hold K=0,1 ; lanes 16-31 hold K=2,3.
    // B layout (4x16 f32): VGPR r: lanes 0-15 row K=r, lanes 16-31 row K=r+2.
    const int nIdx = lane & 15;
    const int hi = lane >> 4;  // 0: lanes 0-15 ; 1: lanes 16-31
    const int bbase = hi * 2 - nIdx + GTAB_BIAS;  // in [1, 18]

    // Hoisted B fragments: identical for all 5 quantities. One aligned
    // ds_load_b64 each; zero padding in the table realizes the band edges.
    v2f bfrag[7];
#pragma unroll
    for (int kc = 0; kc < 7; ++kc) {
      const float2 t = sGP[bbase + 4 * kc];  // rows 4kc+2hi, 4kc+2hi+1
      bfrag[kc].x = t.x;
      bfrag[kc].y = t.y;
    }

    v8f O[NQ];
#pragma unroll
    for (int q = 0; q < NQ; ++q) {
      v8f acc = {0.f, 0.f, 0.f, 0.f, 0.f, 0.f, 0.f, 0.f};
#pragma unroll
      for (int kc = 0; kc < 7; ++kc) {
        const int k0 = kc * 4 + hi * 2;  // even -> 8B-aligned ds_load_b64
        const v2f a = *(const v2f*)&sP[wv][q][nIdx][k0];
        acc = __builtin_amdgcn_wmma_f32_16x16x4_f32(
            false, a, false, bfrag[kc], (short)0, acc, false, false);
      }
      O[q] = acc;
    }

    // SSIM map on the C/D accumulator layout (lane holds 8 row-elements)
    const float c1 = 1.0e-4f;  // (0.01*range)^2
    const float c2 = 9.0e-4f;  // (0.03*range)^2
    float ssum = 0.f;
#pragma unroll
    for (int r = 0; r < 8; ++r) {
      const float mx = O[0][r], my = O[1][r];
      const float exx = O[2][r], eyy = O[3][r], exy = O[4][r];
      const float mxs = mx * mx, mys = my * my, mxy = mx * my;
      const float sxx = exx - mxs, syy = eyy - mys, sxy = exy - mxy;
      const float num = (2.f * mxy + c1) * (2.f * sxy + c2);
      const float den = (mxs + mys + c1) * (sxx + syy + c2);
      ssum += num / den;
    }
#pragma unroll
    for (int off = 16; off > 0; off >>= 1) ssum += __shfl_xor(ssum, off, 32);
    if (lane == 0) ssim_part[tile] = ssum;

    if (s == 0) {
      float msum = 0.f;
      for (int idx = lane; idx < 256; idx += 32) {
        const int m = idx >> 4, n = idx & 15;
        const float d = sX[wv][m + 5][n + 5] - sY[wv][m + 5][n + 5];
        msum += d * d;
      }
#pragma unroll
      for (int off = 16; off > 0; off >>= 1) msum += __shfl_xor(msum, off, 32);
      if (lane == 0) mse_part[tile] = msum;
    }
  }
}

__global__ __launch_bounds__(256)
void finalize_kernel(const float* __restrict__ ssim_part,
                     const float* __restrict__ mse_part,
                     const float* __restrict__ bpp,
                     float* __restrict__ out) {
  __shared__ float red[256];
  __shared__ float totals[6];
  const int tid = threadIdx.x;
  const int segOff[6] = {0, 49152, 61440, 64512, 65280, 0};
  const int segLen[6] = {49152, 12288, 3072, 768, 192, TILES_S0};

  for (int seg = 0; seg < 6; ++seg) {
    const float* base = (seg < 5) ? (ssim_part + segOff[seg]) : mse_part;
    const int len = segLen[seg];
    float acc = 0.f;
    for (int i = tid; i < len; i += 256) acc += base[i];
    red[tid] = acc;
    __syncthreads();
    for (int st = 128; st > 0; st >>= 1) {
      if (tid < st) red[tid] += red[tid + st];
      __syncthreads();
    }
    if (tid == 0) totals[seg] = red[0];
    __syncthreads();
  }

  if (tid == 0) {
    const float wgts[5] = {0.0448f, 0.2856f, 0.3001f, 0.2363f, 0.1333f};
    float scores[5];
    for (int s = 0; s < 5; ++s) {
      const int hw = (512 >> s) * (512 >> s);
      const float npix = 48.f * (float)hw;  // N*C*H*W
      const float m = totals[s] / npix;
      scores[s] = fminf(fmaxf(m, 0.f), 1.f);
    }
    float ms = 1.f;
    for (int s = 0; s < 5; ++s) ms *= powf(fmaxf(scores[s], 1e-6f), wgts[s]);
    ms = fminf(fmaxf(ms, 0.f), 1.f);
    const float mse = totals[5] / (48.f * 512.f * 512.f);
    const float ssim_v = scores[0];
    const float rate = 0.01f * bpp[0];
    const float ssim_loss = 1.f - ssim_v;
    const float ms_loss = 1.f - ms;
    const float total = 1.0f * mse + 0.1f * ssim_loss + 0.1f * ms_loss + rate;
    out[0] = total;
    out[1] = mse;
    out[2] = ssim_loss;
    out[3] = ms_loss;
    out[4] = rate;
    out[5] = ssim_v;
    out[6] = ms;
  }
}

extern "C" void kernel_launch(void* const* d_in, const int* in_sizes, int n_in,
                              void* d_out, int out_size, void* d_ws,
                              size_t ws_size, hipStream_t stream) {
  const float* pred = (const float*)d_in[0];
  const float* tgt = (const float*)d_in[1];
  const float* bpp = (const float*)d_in[2];
  float* out = (float*)d_out;

  float* ssim_part = (float*)d_ws;              // TILES_TOTAL floats
  float* mse_part = ssim_part + TILES_TOTAL;    // TILES_S0 floats (~458 KB total)

  const int blocks = (TILES_TOTAL + WAVES_PER_BLOCK - 1) / WAVES_PER_BLOCK;
  ssim_tile_kernel<<<blocks, WAVES_PER_BLOCK * 32, 0, stream>>>(
      pred, tgt, ssim_part, mse_part);
  finalize_kernel<<<1, 256, 0, stream>>>(ssim_part, mse_part, bpp, out);
}